// ms_ssimloss_90494960927405
// MI455X (gfx1250) — compile-verified
//
#include <hip/hip_runtime.h>

typedef float v2f __attribute__((ext_vector_type(2)));
typedef float v8f __attribute__((ext_vector_type(8)));

#define C1F 1.0e-4f
#define C2F 9.0e-4f

// Canonical 11-tap Gaussian window, sigma=1.5 (normalized).
static __device__ __constant__ float GW[11] = {
    0.001028380f, 0.007598758f, 0.036000773f, 0.109360762f,
    0.213005121f, 0.266011737f, 0.213005121f, 0.109360762f,
    0.036000773f, 0.007598758f, 0.001028380f};

#define FR  32   // field rows (26 valid + zero pad)
#define FCS 30   // field row stride in floats (28 cols used + pad)
#define HCS 17   // Hout row stride (16 cols + pad to dodge LDS bank conflicts)

// One wave (32 lanes) per 16x16 output tile. Fully fused: load halo tile once,
// run 5 separable Gaussian convs as banded-matrix f32 WMMAs, pointwise SSIM,
// tile-sum -> partials, AND emit the 2x2-avg-pooled 8x8 tile for the next
// pyramid level straight from LDS (no second read of the images).
__global__ void __launch_bounds__(32)
ssim_tile_wmma(const float* __restrict__ img1,
               const float* __restrict__ img2,
               float* __restrict__ partials,
               float* __restrict__ pool1,   // next-level img1 (nullptr on last level)
               float* __restrict__ pool2,   // next-level img2
               int D, int tilesX)
{
    __shared__ float F0[FR * FCS];
    __shared__ float F1[FR * FCS];
    __shared__ float FT[FR * FCS];
    __shared__ float HB[FR * HCS];

    const int lane = threadIdx.x;
    const int m    = lane & 15;          // M (or N) index of this lane
    const int koff = (lane >> 4) << 1;   // K offset for upper half-wave

    const int tile  = blockIdx.x;
    const int plane = blockIdx.y;
    const int ty = tile / tilesX;
    const int tx = tile - ty * tilesX;

    const float* __restrict__ p1 = img1 + (size_t)plane * D * D;
    const float* __restrict__ p2 = img2 + (size_t)plane * D * D;

    // Zero staging buffers (covers all pad rows/cols -> implicit zero padding).
    for (int i = lane; i < FR * FCS; i += 32) { F0[i] = 0.0f; F1[i] = 0.0f; }
    __syncthreads();

    // Load the 26x26 halo tile (rows/cols -5..+20 of the output tile).
    const int baseR = ty * 16 - 5;
    const int baseC = tx * 16 - 5;
    for (int r = 0; r < 26; ++r) {
        const int gr = baseR + r;
        if (lane < 26) {
            const int gc = baseC + lane;
            float a = 0.0f, b = 0.0f;
            if (gr >= 0 && gr < D && gc >= 0 && gc < D) {
                const size_t o = (size_t)gr * D + gc;
                a = p1[o];
                b = p2[o];
            }
            F0[r * FCS + lane] = a;
            F1[r * FCS + lane] = b;
        }
    }
    __syncthreads();

    // Fused 2x2 avg-pool of the tile interior (F rows/cols 5..20 == the exact
    // 16x16 image tile). Each lane produces 2 adjacent pooled outputs per
    // image; stores drain (STOREcnt) while the WMMA pipeline below runs.
    if (pool1 != nullptr) {
        const int Do = D >> 1;
        const int pr = lane >> 2;          // 0..7 pooled row
        const int pc = (lane & 3) << 1;    // 0,2,4,6 pooled col pair
        float2 ra, rb;
#pragma unroll
        for (int j = 0; j < 2; ++j) {
            const int fr = 5 + 2 * pr;
            const int fc = 5 + 2 * (pc + j);
            const float a = 0.25f * (F0[fr * FCS + fc]       + F0[fr * FCS + fc + 1] +
                                     F0[(fr + 1) * FCS + fc] + F0[(fr + 1) * FCS + fc + 1]);
            const float b = 0.25f * (F1[fr * FCS + fc]       + F1[fr * FCS + fc + 1] +
                                     F1[(fr + 1) * FCS + fc] + F1[(fr + 1) * FCS + fc + 1]);
            if (j == 0) { ra.x = a; rb.x = b; } else { ra.y = a; rb.y = b; }
        }
        const size_t obase = (size_t)plane * Do * Do +
                             (size_t)(ty * 8 + pr) * Do + (tx * 8 + pc);
        *(float2*)(pool1 + obase) = ra;
        *(float2*)(pool2 + obase) = rb;
    }

    // Constant banded Gaussian fragments: value g[K - m] for K-m in [0,10].
    // 32-bit A layout: A[M=lane%16][K = v + 2*(lane/16)]; the B layout uses
    // the mirrored mapping B[K = v + 2*(lane/16)][N=lane%16], so one fragment
    // set serves as B (horizontal: Wh[k][j]=g[k-j]) and A (vertical:
    // Wv[i][k]=g[k-i]).
    v2f wf[7];
#pragma unroll
    for (int kk = 0; kk < 7; ++kk) {
#pragma unroll
        for (int v = 0; v < 2; ++v) {
            const int K = 4 * kk + koff + v;
            const int d = K - m;
            wf[kk][v] = (d >= 0 && d <= 10) ? GW[d] : 0.0f;
        }
    }

    v8f acc[5];  // conv(i1), conv(i2), conv(i1*i1), conv(i2*i2), conv(i1*i2)

#pragma unroll
    for (int f = 0; f < 5; ++f) {
        const float* Fin;
        if (f == 0) Fin = F0;
        else if (f == 1) Fin = F1;
        else {
            for (int i = lane; i < FR * FCS; i += 32) {
                const float a = F0[i], b = F1[i];
                FT[i] = (f == 2) ? a * a : (f == 3) ? b * b : a * b;
            }
            __syncthreads();
            Fin = FT;
        }

        // Horizontal: Hout(32x16) = Fin(32x28) x Wh(28x16), 2 M-tiles.
#pragma unroll
        for (int mt = 0; mt < 2; ++mt) {
            v8f h = {};
#pragma unroll
            for (int kk = 0; kk < 7; ++kk) {
                const int row = mt * 16 + m;
                const int col = 4 * kk + koff;
                v2f a;
                a.x = Fin[row * FCS + col];
                a.y = Fin[row * FCS + col + 1];
                h = __builtin_amdgcn_wmma_f32_16x16x4_f32(
                        false, a, false, wf[kk], (short)0, h, false, false);
            }
            // D layout: D[M = v + 8*(lane/16)][N = lane%16]
#pragma unroll
            for (int v = 0; v < 8; ++v) {
                const int row = mt * 16 + v + ((lane >> 4) << 3);
                HB[row * HCS + m] = h[v];
            }
        }
        __syncthreads();

        // Vertical: Out(16x16) = Wv(16x28) x Hout(28x16).
        v8f o = {};
#pragma unroll
        for (int kk = 0; kk < 7; ++kk) {
            const int K0 = 4 * kk + koff;
            v2f b;
            b.x = HB[K0 * HCS + m];
            b.y = HB[(K0 + 1) * HCS + m];
            o = __builtin_amdgcn_wmma_f32_16x16x4_f32(
                    false, wf[kk], false, b, (short)0, o, false, false);
        }
        acc[f] = o;
        __syncthreads();
    }

    // Pointwise SSIM + tile sum (layout-agnostic: all accs share C/D layout).
    float tsum = 0.0f;
#pragma unroll
    for (int v = 0; v < 8; ++v) {
        const float m1 = acc[0][v], m2 = acc[1][v];
        const float e11 = acc[2][v], e22 = acc[3][v], e12 = acc[4][v];
        const float m1sq = m1 * m1, m2sq = m2 * m2, m12 = m1 * m2;
        const float s1 = e11 - m1sq, s2 = e22 - m2sq, s12 = e12 - m12;
        const float num = (2.0f * m12 + C1F) * (2.0f * s12 + C2F);
        const float den = (m1sq + m2sq + C1F) * (s1 + s2 + C2F);
        tsum += num / den;
    }
#pragma unroll
    for (int off = 16; off > 0; off >>= 1)
        tsum += __shfl_xor(tsum, off, 32);
    if (lane == 0)
        partials[(size_t)blockIdx.y * gridDim.x + blockIdx.x] = tsum;
}

// Deterministic fixed-order reduction of one level's tile partials.
// 1024 threads (32 waves on one WGP) to shorten the strided phase.
__global__ void __launch_bounds__(1024)
reduce_level(const float* __restrict__ partials, int n,
             float* __restrict__ slot, float invN)
{
    __shared__ float sm[1024];
    float acc = 0.0f;
    for (int i = threadIdx.x; i < n; i += 1024) acc += partials[i];
    sm[threadIdx.x] = acc;
    __syncthreads();
    for (int s = 512; s > 0; s >>= 1) {
        if (threadIdx.x < s) sm[threadIdx.x] += sm[threadIdx.x + s];
        __syncthreads();
    }
    if (threadIdx.x == 0) *slot = 1.0f - sm[0] * invN;
}

__global__ void finalize5(const float* __restrict__ slots, float* __restrict__ out)
{
    if (threadIdx.x == 0) {
        float t = 0.0f;
        for (int i = 0; i < 5; ++i) t += slots[i];
        out[0] = t;
    }
}

extern "C" void kernel_launch(void* const* d_in, const int* in_sizes, int n_in,
                              void* d_out, int out_size, void* d_ws, size_t ws_size,
                              hipStream_t stream) {
    (void)in_sizes; (void)n_in; (void)out_size; (void)ws_size;
    const float* i1 = (const float*)d_in[0];
    const float* i2 = (const float*)d_in[1];
    float* out = (float*)d_out;
    float* ws  = (float*)d_ws;

    const int planes = 16 * 3;
    const int dims[5] = {512, 256, 128, 64, 32};

    const float* b1[5]; const float* b2[5];
    float* w1[5]; float* w2[5];
    b1[0] = i1; b2[0] = i2; w1[0] = nullptr; w2[0] = nullptr;
    size_t off = 0;
    for (int l = 1; l < 5; ++l) {
        const size_t n = (size_t)planes * dims[l] * dims[l];
        w1[l] = ws + off; off += n;
        w2[l] = ws + off; off += n;
        b1[l] = w1[l]; b2[l] = w2[l];
    }
    float* partials = ws + off; off += (size_t)planes * 32 * 32;
    float* slots    = ws + off; off += 8;

    for (int l = 0; l < 5; ++l) {
        const int d = dims[l];
        const int tilesX = d / 16;
        const int nt = tilesX * tilesX;
        float* p1 = (l < 4) ? w1[l + 1] : nullptr;
        float* p2 = (l < 4) ? w2[l + 1] : nullptr;
        ssim_tile_wmma<<<dim3(nt, planes), 32, 0, stream>>>(
            b1[l], b2[l], partials, p1, p2, d, tilesX);
        reduce_level<<<1, 1024, 0, stream>>>(
            partials, nt * planes, slots + l, 1.0f / ((float)planes * d * d));
    }
    finalize5<<<1, 32, 0, stream>>>(slots, out);
}